// LabelSmoothSoftmaxCEV1_68393059221597
// MI455X (gfx1250) — compile-verified
//
#include <hip/hip_runtime.h>
#include <hip/hip_bf16.h>
#include <stdint.h>

// Problem constants (fixed by setup_inputs()).
#define NCLS   19
#define HW     (512 * 512)            // pixels per image
#define NPIX   (8 * 512 * 512)        // total pixels
#define TPB    256                    // threads per block
#define PPT    4                      // pixels per thread (float4 path)
#define PPB    (TPB * PPT)            // 1024 pixels per block
#define NBLK   (NPIX / PPB)           // 2048 blocks

#define IGNORE_IDX 255

// Label-smoothing constants (match reference float math).
__device__ __constant__ float kBeta  = 0.1f / 19.0f;                   // lb_neg
__device__ __constant__ float kAlpha = (1.0f - 0.1f) - (0.1f / 19.0f); // lb_pos - lb_neg
__device__ __constant__ float kGamma = ((1.0f - 0.1f) - (0.1f / 19.0f)) + (0.1f / 19.0f) * 19.0f;

__global__ void lsce_init_ws(double* ws) {
    if (threadIdx.x == 0) { ws[0] = 0.0; ws[1] = 0.0; }
}

__global__ __launch_bounds__(TPB)
void lsce_main(const float* __restrict__ logits,
               const int*   __restrict__ label,
               double*      __restrict__ ws) {
    // LDS tile: 19 classes x 1024 pixels of fp32 = 76 KB, staged by the TDM-era
    // async-to-LDS path (gfx1250 GLOBAL_LOAD_ASYNC_TO_LDS_B128, ASYNCcnt).
    __shared__ float4 tile4[NCLS][TPB];
    __shared__ float  red[TPB];
    __shared__ float  redc[TPB];

    const int tid = threadIdx.x;
    const long long blockPix = (long long)blockIdx.x * PPB;   // HW % PPB == 0 -> one image per block
    const int n = (int)(blockPix >> 18);                      // / HW (2^18)
    const int r = (int)(blockPix & (HW - 1));

    const float* gbase = logits + (size_t)n * NCLS * HW + r + tid * PPT;

    // --- Stage logits tile into LDS via async DMA (one b128 per lane per class) ---
#pragma unroll
    for (int c = 0; c < NCLS; ++c) {
        unsigned lds_addr = (unsigned)(uintptr_t)&tile4[c][tid];          // low 32b of shared ptr == LDS byte offset
        unsigned long long gaddr = (unsigned long long)(uintptr_t)(gbase + (size_t)c * HW);
        asm volatile("global_load_async_to_lds_b128 %0, %1, off"
                     :: "v"(lds_addr), "v"(gaddr)
                     : "memory");
    }
    asm volatile("s_wait_asynccnt 0x0" ::: "memory");   // this wave's async writes landed in LDS
    __syncthreads();                                    // all waves' writes visible

    // --- Labels (int32 per harness convention), 4 per thread ---
    const int4 lv = *(const int4*)(label + blockPix + tid * PPT);
    int t[PPT] = { lv.x, lv.y, lv.z, lv.w };

    // --- Pass 1: per-pixel max and sum of logits (ds_load_b128 per class) ---
    float m[PPT], sx[PPT];
#pragma unroll
    for (int j = 0; j < PPT; ++j) { m[j] = -3.402823466e+38f; sx[j] = 0.0f; }

#pragma unroll
    for (int c = 0; c < NCLS; ++c) {
        const float4 v = tile4[c][tid];
        m[0] = fmaxf(m[0], v.x); sx[0] += v.x;
        m[1] = fmaxf(m[1], v.y); sx[1] += v.y;
        m[2] = fmaxf(m[2], v.z); sx[2] += v.z;
        m[3] = fmaxf(m[3], v.w); sx[3] += v.w;
    }

    // --- Pass 2: sum of exp(x - max), one v_exp_f32 per element ---
    float s[PPT] = { 0.0f, 0.0f, 0.0f, 0.0f };
#pragma unroll
    for (int c = 0; c < NCLS; ++c) {
        const float4 v = tile4[c][tid];
        s[0] += __expf(v.x - m[0]);
        s[1] += __expf(v.y - m[1]);
        s[2] += __expf(v.z - m[2]);
        s[3] += __expf(v.w - m[3]);
    }

    // --- Per-pixel loss ---
    const float* tileS = (const float*)tile4;   // [c][pix] flat: c*PPB + pix
    float lsum = 0.0f, lcnt = 0.0f;
#pragma unroll
    for (int j = 0; j < PPT; ++j) {
        const bool ign = (t[j] == IGNORE_IDX);
        const int  tc  = ign ? 0 : t[j];
        const float xt  = tileS[tc * PPB + tid * PPT + j];
        const float lse = m[j] + __logf(s[j]);
        const float L   = -(kAlpha * xt + kBeta * sx[j] - kGamma * lse);
        if (!ign) { lsum += L; lcnt += 1.0f; }
    }

    // --- Block tree reduction in LDS, then one f64 atomic pair per block ---
    red[tid] = lsum; redc[tid] = lcnt;
    __syncthreads();
#pragma unroll
    for (int off = TPB / 2; off > 0; off >>= 1) {
        if (tid < off) { red[tid] += red[tid + off]; redc[tid] += redc[tid + off]; }
        __syncthreads();
    }
    if (tid == 0) {
        atomicAdd(&ws[0], (double)red[0]);
        atomicAdd(&ws[1], (double)redc[0]);
    }
}

__global__ void lsce_finalize(const double* __restrict__ ws, float* __restrict__ out) {
    if (threadIdx.x == 0) out[0] = (float)(ws[0] / ws[1]);
}

extern "C" void kernel_launch(void* const* d_in, const int* in_sizes, int n_in,
                              void* d_out, int out_size, void* d_ws, size_t ws_size,
                              hipStream_t stream) {
    const float* logits = (const float*)d_in[0];
    const int*   label  = (const int*)d_in[1];
    double*      ws     = (double*)d_ws;
    float*       out    = (float*)d_out;

    lsce_init_ws<<<1, 32, 0, stream>>>(ws);
    lsce_main<<<NBLK, TPB, 0, stream>>>(logits, label, ws);
    lsce_finalize<<<1, 32, 0, stream>>>(ws, out);
}